// GNN21_27410481283390
// MI455X (gfx1250) — compile-verified
//
#include <hip/hip_runtime.h>
#include <hip/hip_bf16.h>
#include <math.h>

// ---------------- problem constants (from reference) ----------------
#define N_NODES 50000
#define N_EDGES 1600000
#define H1 6
#define F1 16
#define H2 6
#define F2 24
#define D_IN 11
#define K1PAD 12       // D_IN padded to multiple of 4 for WMMA K-steps
#define HF1 (H1*F1)    // 96
#define HF2 (H2*F2)    // 144
#define NEG_SLOPE 0.2f
#define EPS_ATT 1e-12f

typedef float v2f __attribute__((ext_vector_type(2)));
typedef float v8f __attribute__((ext_vector_type(8)));

// ---------------- fp32 WMMA GEMM: D[M x N] = A[M x K] * B[K x N] ----------------
// One wave (32 threads) computes one 16x16 tile of D with V_WMMA_F32_16X16X4_F32.
// All shapes compile-time: K%4==0, M%16==0 (50000 = 3125*16), no guards anywhere.
// ISA 7.12.2 layouts:
//   A 16x4 : lane l -> row m=l&15 ; VGPR v holds K = k0 + v + (l>=16 ? 2 : 0)
//   B 4x16 : lane l -> col n=l&15 ; VGPR v holds K = k0 + v + (l>=16 ? 2 : 0)
//   C/D    : lane l -> col n=l&15 ; VGPR r holds row m = r + (l>=16 ? 8 : 0)
template<int K, int LDA, int LDB, int LDC>
__global__ void gemm_wmma_f32(const float* __restrict__ A, const float* __restrict__ B,
                              float* __restrict__ D, int ntiles)
{
    const int tile  = blockIdx.x;
    const int mtile = tile / ntiles;
    const int ntile = tile - mtile * ntiles;
    const int lane  = threadIdx.x & 31;
    const int l15   = lane & 15;
    const int khalf = (lane >> 4) << 1;       // 0 or 2

    // Pre-offset base pointers so the unrolled loop uses immediate offsets only.
    const float* __restrict__ ap = A + (size_t)(mtile * 16 + l15) * LDA + khalf;
    const float* __restrict__ bp = B + (size_t)khalf * LDB + ntile * 16 + l15;

    v8f acc = {};
#pragma unroll
    for (int k0 = 0; k0 < K; k0 += 4) {
        v2f a = *(const v2f*)(ap + k0);       // 8B-aligned: LDA and khalf are even
        v2f b;
        b.x = bp[(size_t)(k0    ) * LDB];
        b.y = bp[(size_t)(k0 + 1) * LDB];
        acc = __builtin_amdgcn_wmma_f32_16x16x4_f32(false, a, false, b,
                                                    (short)0, acc, false, false);
    }

    float* __restrict__ dp = D + (size_t)(mtile * 16 + ((lane >> 4) * 8)) * LDC
                               + ntile * 16 + l15;
#pragma unroll
    for (int r = 0; r < 8; ++r)
        dp[(size_t)r * LDC] = acc[r];
}

// ---------------- K-padding staging for layer 1 ----------------
__global__ void pad_x_kernel(const float* __restrict__ x, float* __restrict__ xp)
{
    int i = blockIdx.x * blockDim.x + threadIdx.x;       // over N_NODES*K1PAD
    if (i >= N_NODES * K1PAD) return;
    int n = i / K1PAD, k = i - n * K1PAD;
    xp[i] = (k < D_IN) ? x[n * D_IN + k] : 0.f;
}

__global__ void pad_w1_kernel(const float* __restrict__ W1, float* __restrict__ W1p)
{
    int i = blockIdx.x * blockDim.x + threadIdx.x;       // over K1PAD*HF1
    if (i >= K1PAD * HF1) return;
    int k = i / HF1, f = i - k * HF1;
    W1p[i] = (k < D_IN) ? W1[k * HF1 + f] : 0.f;
}

// ---------------- per-node attention coefficients ----------------
template<int H, int F>
__global__ void alpha_kernel(const float* __restrict__ h,
                             const float* __restrict__ a_src,
                             const float* __restrict__ a_dst,
                             float* __restrict__ as_n, float* __restrict__ ad_n)
{
    int idx = blockIdx.x * blockDim.x + threadIdx.x;
    if (idx >= N_NODES * H) return;
    int n  = idx / H;
    int hh = idx - n * H;
    const float* hp = h + (size_t)n * (H * F) + hh * F;
    float ss = 0.f, sd = 0.f;
#pragma unroll
    for (int f = 0; f < F; ++f) {
        float v = hp[f];
        ss = fmaf(v, a_src[hh * F + f], ss);
        sd = fmaf(v, a_dst[hh * F + f], sd);
    }
    as_n[idx] = ss;
    ad_n[idx] = sd;
}

// ---------------- edge-phase helpers ----------------
__device__ __forceinline__ float leaky(float x) { return x > 0.f ? x : NEG_SLOPE * x; }

// float atomic max via signed-int (positives) / unsigned-int (negatives) ordering.
__device__ __forceinline__ void atomicMaxF(float* addr, float val)
{
    if (val >= 0.f) atomicMax((int*)addr,          __float_as_int(val));
    else            atomicMin((unsigned int*)addr, __float_as_uint(val));
}

template<int H>
__global__ void edge_max_kernel(const int* __restrict__ src, const int* __restrict__ dst,
                                const float* __restrict__ as_n, const float* __restrict__ ad_n,
                                float* __restrict__ m)
{
    int idx = blockIdx.x * blockDim.x + threadIdx.x;
    if (idx >= N_EDGES * H) return;
    int e  = idx / H;
    int hh = idx - e * H;
    int s = src[e], d = dst[e];
    float ev = leaky(as_n[s * H + hh] + ad_n[d * H + hh]);
    atomicMaxF(&m[d * H + hh], ev);
}

template<int H>
__global__ void edge_denom_kernel(const int* __restrict__ src, const int* __restrict__ dst,
                                  const float* __restrict__ as_n, const float* __restrict__ ad_n,
                                  const float* __restrict__ m, float* __restrict__ denom)
{
    int idx = blockIdx.x * blockDim.x + threadIdx.x;
    if (idx >= N_EDGES * H) return;
    int e  = idx / H;
    int hh = idx - e * H;
    int s = src[e], d = dst[e];
    float ev = leaky(as_n[s * H + hh] + ad_n[d * H + hh]);
    float mm = m[d * H + hh];
    if (!(mm > -1e38f && mm < 1e38f)) mm = 0.f;     // mirror where(isfinite(m), m, 0)
    atomicAdd(&denom[d * H + hh], __expf(ev - mm));
}

template<int H, int F>
__global__ void edge_msg_kernel(const int* __restrict__ src, const int* __restrict__ dst,
                                const float* __restrict__ as_n, const float* __restrict__ ad_n,
                                const float* __restrict__ m, const float* __restrict__ denom,
                                const float* __restrict__ h, float* __restrict__ out)
{
    int idx = blockIdx.x * blockDim.x + threadIdx.x;
    if (idx >= N_EDGES * H) return;
    int e  = idx / H;
    int hh = idx - e * H;
    int s = src[e], d = dst[e];
    const float* hs = h + (size_t)s * (H * F) + hh * F;
    __builtin_prefetch(hs, 0, 1);                    // global_prefetch_b8 on the gathered row
    float ev = leaky(as_n[s * H + hh] + ad_n[d * H + hh]);
    float mm = m[d * H + hh];
    if (!(mm > -1e38f && mm < 1e38f)) mm = 0.f;
    float att = __expf(ev - mm) / (denom[d * H + hh] + EPS_ATT);
    float* od = out + (size_t)d * (H * F) + hh * F;
#pragma unroll
    for (int f = 0; f < F; ++f)
        atomicAdd(&od[f], hs[f] * att);
}

// ---------------- elementwise / init / reduce ----------------
__global__ void elu_kernel(float* __restrict__ x, int n)
{
    int i = blockIdx.x * blockDim.x + threadIdx.x;
    if (i < n) {
        float v = x[i];
        x[i] = v > 0.f ? v : (__expf(v) - 1.f);
    }
}

__global__ void fill_f32_kernel(float* __restrict__ p, float v, int n)
{
    int i = blockIdx.x * blockDim.x + threadIdx.x;
    if (i < n) p[i] = v;
}

__global__ void fill_bits_kernel(unsigned int* __restrict__ p, unsigned int v, int n)
{
    int i = blockIdx.x * blockDim.x + threadIdx.x;
    if (i < n) p[i] = v;
}

__global__ void pool_kernel(const float* __restrict__ h2, float* __restrict__ pooled)
{
    __shared__ float s[HF2];
    for (int f = threadIdx.x; f < HF2; f += blockDim.x) s[f] = 0.f;
    __syncthreads();
    const int total = N_NODES * HF2;
    for (int i = blockIdx.x * blockDim.x + threadIdx.x; i < total; i += gridDim.x * blockDim.x) {
        int f = i % HF2;
        atomicAdd(&s[f], h2[i]);
    }
    __syncthreads();
    for (int f = threadIdx.x; f < HF2; f += blockDim.x)
        atomicAdd(&pooled[f], s[f]);
}

__global__ void final_kernel(const float* __restrict__ pooled, const float* __restrict__ Wd,
                             const float* __restrict__ bd, float* __restrict__ out)
{
    if (blockIdx.x == 0 && threadIdx.x == 0) {
        float ss = 0.f;
        for (int f = 0; f < HF2; ++f) ss += pooled[f] * pooled[f];
        float norm = fmaxf(sqrtf(ss), 1e-12f);
        float acc = 0.f;
        for (int f = 0; f < HF2; ++f) acc += (pooled[f] / norm) * Wd[f];
        out[0] = acc + bd[0];
    }
}

// ---------------- launch ----------------
extern "C" void kernel_launch(void* const* d_in, const int* in_sizes, int n_in,
                              void* d_out, int out_size, void* d_ws, size_t ws_size,
                              hipStream_t stream)
{
    const float* x   = (const float*)d_in[0];
    const int*   src = (const int*)  d_in[1];
    const int*   dst = (const int*)  d_in[2];
    const float* W1  = (const float*)d_in[3];
    const float* a1s = (const float*)d_in[4];
    const float* a1d = (const float*)d_in[5];
    const float* W2  = (const float*)d_in[6];
    const float* a2s = (const float*)d_in[7];
    const float* a2d = (const float*)d_in[8];
    const float* Wd  = (const float*)d_in[9];
    const float* bd  = (const float*)d_in[10];
    float* out = (float*)d_out;

    // carve workspace (~109 MB total)
    char* ws = (char*)d_ws;
    size_t off = 0;
    auto alloc_f = [&](size_t nelem) -> float* {
        float* p = (float*)(ws + off);
        off = (off + nelem * sizeof(float) + 255) & ~(size_t)255;
        return p;
    };
    float* h1    = alloc_f((size_t)N_NODES * HF1);   // xpad @ W1pad
    float* h1agg = alloc_f((size_t)N_NODES * HF1);   // layer-1 aggregated (then ELU'd)
    float* h2lin = alloc_f((size_t)N_NODES * HF2);   // h1act @ W2
    float* h2agg = alloc_f((size_t)N_NODES * HF2);   // layer-2 aggregated (then ELU'd)
    float* xpad  = alloc_f((size_t)N_NODES * K1PAD); // x zero-padded K: 11 -> 12
    float* W1pad = alloc_f((size_t)K1PAD * HF1);     // W1 zero-padded K: 11 -> 12
    float* a1sN  = alloc_f((size_t)N_NODES * H1);
    float* a1dN  = alloc_f((size_t)N_NODES * H1);
    float* a2sN  = alloc_f((size_t)N_NODES * H2);
    float* a2dN  = alloc_f((size_t)N_NODES * H2);
    float* m1    = alloc_f((size_t)N_NODES * H1);
    float* den1  = alloc_f((size_t)N_NODES * H1);
    float* m2    = alloc_f((size_t)N_NODES * H2);
    float* den2  = alloc_f((size_t)N_NODES * H2);
    float* pooled = alloc_f(HF2);

    const int TPB = 256;
    const unsigned NEG_INF_BITS = 0xFF800000u;
    const int NH   = N_NODES * H1;               // == N_NODES*H2
    const int EH   = N_EDGES * H1;               // == N_EDGES*H2
    const int nhG  = (NH + TPB - 1) / TPB;
    const int ehG  = (EH + TPB - 1) / TPB;
    const int n96  = N_NODES * HF1;
    const int n144 = N_NODES * HF2;

    // ---- init scratch ----
    fill_bits_kernel<<<nhG, TPB, 0, stream>>>((unsigned*)m1, NEG_INF_BITS, NH);
    fill_bits_kernel<<<nhG, TPB, 0, stream>>>((unsigned*)m2, NEG_INF_BITS, NH);
    fill_f32_kernel<<<nhG, TPB, 0, stream>>>(den1, 0.f, NH);
    fill_f32_kernel<<<nhG, TPB, 0, stream>>>(den2, 0.f, NH);
    fill_f32_kernel<<<(n96  + TPB - 1) / TPB, TPB, 0, stream>>>(h1agg, 0.f, n96);
    fill_f32_kernel<<<(n144 + TPB - 1) / TPB, TPB, 0, stream>>>(h2agg, 0.f, n144);
    fill_f32_kernel<<<1, TPB, 0, stream>>>(pooled, 0.f, HF2);

    const int MTILES = N_NODES / 16;             // 3125, exact

    // ---- layer 1 ----
    pad_x_kernel<<<(N_NODES * K1PAD + TPB - 1) / TPB, TPB, 0, stream>>>(x, xpad);
    pad_w1_kernel<<<(K1PAD * HF1 + TPB - 1) / TPB, TPB, 0, stream>>>(W1, W1pad);
    gemm_wmma_f32<K1PAD, K1PAD, HF1, HF1>
        <<<MTILES * (HF1 / 16), 32, 0, stream>>>(xpad, W1pad, h1, HF1 / 16);
    alpha_kernel<H1, F1><<<nhG, TPB, 0, stream>>>(h1, a1s, a1d, a1sN, a1dN);
    edge_max_kernel<H1><<<ehG, TPB, 0, stream>>>(src, dst, a1sN, a1dN, m1);
    edge_denom_kernel<H1><<<ehG, TPB, 0, stream>>>(src, dst, a1sN, a1dN, m1, den1);
    edge_msg_kernel<H1, F1><<<ehG, TPB, 0, stream>>>(src, dst, a1sN, a1dN, m1, den1, h1, h1agg);
    elu_kernel<<<(n96 + TPB - 1) / TPB, TPB, 0, stream>>>(h1agg, n96);

    // ---- layer 2 ----
    gemm_wmma_f32<HF1, HF1, HF2, HF2>
        <<<MTILES * (HF2 / 16), 32, 0, stream>>>(h1agg, W2, h2lin, HF2 / 16);
    alpha_kernel<H2, F2><<<nhG, TPB, 0, stream>>>(h2lin, a2s, a2d, a2sN, a2dN);
    edge_max_kernel<H2><<<ehG, TPB, 0, stream>>>(src, dst, a2sN, a2dN, m2);
    edge_denom_kernel<H2><<<ehG, TPB, 0, stream>>>(src, dst, a2sN, a2dN, m2, den2);
    edge_msg_kernel<H2, F2><<<ehG, TPB, 0, stream>>>(src, dst, a2sN, a2dN, m2, den2, h2lin, h2agg);
    elu_kernel<<<(n144 + TPB - 1) / TPB, TPB, 0, stream>>>(h2agg, n144);

    // ---- pool + normalize + linear head ----
    pool_kernel<<<1024, TPB, 0, stream>>>(h2agg, pooled);
    final_kernel<<<1, 32, 0, stream>>>(pooled, Wd, bd, out);
}